// SpecLayer_80418967650840
// MI455X (gfx1250) — compile-verified
//
#include <hip/hip_runtime.h>
#include <hip/hip_bf16.h>

typedef float v2f __attribute__((ext_vector_type(2)));
typedef float v8f __attribute__((ext_vector_type(8)));

constexpr int GG = 128;
constexpr int NN = 1024;
constexpr int EE = 64;
constexpr int II = 128;
constexpr int HH = 128;
constexpr int OO = 128;

__device__ __forceinline__ v8f wmma_f32(v2f a, v2f b, v8f c) {
    // D = A(16x4 f32) * B(4x16 f32) + C(16x16 f32)
    return __builtin_amdgcn_wmma_f32_16x16x4_f32(
        /*neg_a=*/false, a, /*neg_b=*/false, b,
        /*c_mod=*/(short)0, c, /*reuse_a=*/false, /*reuse_b=*/false);
}

__device__ __forceinline__ v2f ld_v2f(const float* p) {
    return *(const v2f*)__builtin_assume_aligned(p, 8);   // -> ds_load_b64
}

// ---------------------------------------------------------------------------
// Kernel 1: spec_in[g,e,i] = sum_n eigen[g,n,e] * feature[g,n,i]
// One WG per graph. Both operands stored K(n)-contiguous (transposed) in LDS.
// Double-buffered K-chunks of 32. Wave w: 2 e-tiles x 2 i-tiles (32x32).
// ---------------------------------------------------------------------------
constexpr int KPAD = 34;                 // 34%32==2 -> loads conflict-free
constexpr int EBUF = 64 * KPAD;          // et: [64][KPAD]
constexpr int FBUF = 128 * KPAD;         // ft: [128][KPAD]
constexpr int ABUF = EBUF + FBUF;        // one double-buffer slot

__device__ __forceinline__ void stage_chunkA(float* Ed, float* Fd, int tid,
                                             const float4* ev, const float4* fv) {
#pragma unroll
    for (int j = 0; j < 2; ++j) {        // eigen 32x64 -> et[e][n]
        const int idx = tid + j * 256;
        const int n = idx >> 4, c4 = idx & 15;
        Ed[(c4 * 4 + 0) * KPAD + n] = ev[j].x;
        Ed[(c4 * 4 + 1) * KPAD + n] = ev[j].y;
        Ed[(c4 * 4 + 2) * KPAD + n] = ev[j].z;
        Ed[(c4 * 4 + 3) * KPAD + n] = ev[j].w;
    }
#pragma unroll
    for (int j = 0; j < 4; ++j) {        // feature 32x128 -> ft[i][n]
        const int idx = tid + j * 256;
        const int n = idx >> 5, c4 = idx & 31;
        Fd[(c4 * 4 + 0) * KPAD + n] = fv[j].x;
        Fd[(c4 * 4 + 1) * KPAD + n] = fv[j].y;
        Fd[(c4 * 4 + 2) * KPAD + n] = fv[j].z;
        Fd[(c4 * 4 + 3) * KPAD + n] = fv[j].w;
    }
}

__global__ __launch_bounds__(256) void k_spec_in(const float* __restrict__ feature,
                                                 const float* __restrict__ eigen,
                                                 float* __restrict__ spec_in) {
    const int g    = blockIdx.x;
    const int tid  = threadIdx.x;
    const int lane = tid & 31;
    const int wave = tid >> 5;

    extern __shared__ float smem[];

    const int mm   = lane & 15;
    const int kb   = (lane >> 4) << 1;
    const int half = lane >> 4;

    const int e0 = (wave & 1) * 32;
    const int i0 = (wave >> 1) * 32;

    const float* egbase = eigen   + (size_t)g * NN * EE;
    const float* fgbase = feature + (size_t)g * NN * II;

    v8f acc[2][2] = {};
    float4 ev[2], fv[4];

    // preload + stage chunk 0
    {
        const float4* esrc = (const float4*)egbase;
        ev[0] = esrc[tid]; ev[1] = esrc[tid + 256];
        const float4* fsrc = (const float4*)fgbase;
        fv[0] = fsrc[tid];       fv[1] = fsrc[tid + 256];
        fv[2] = fsrc[tid + 512]; fv[3] = fsrc[tid + 768];
        stage_chunkA(smem, smem + EBUF, tid, ev, fv);
    }
    __syncthreads();

    constexpr int NCH = NN / 32;
    for (int c = 0; c < NCH; ++c) {
        const int cur = c & 1;
        const bool more = (c + 1) < NCH;
        if (more) {   // issue next chunk's global loads up front (latency hiding)
            const float4* esrc = (const float4*)(egbase + (size_t)(c + 1) * 32 * EE);
            ev[0] = esrc[tid]; ev[1] = esrc[tid + 256];
            const float4* fsrc = (const float4*)(fgbase + (size_t)(c + 1) * 32 * II);
            fv[0] = fsrc[tid];       fv[1] = fsrc[tid + 256];
            fv[2] = fsrc[tid + 512]; fv[3] = fsrc[tid + 768];
        }
        const float* Et = smem + cur * ABUF;
        const float* Ft = Et + EBUF;
#pragma unroll
        for (int ks = 0; ks < 8; ++ks) {
            const int k4 = ks * 4;
            v2f a0 = ld_v2f(&Et[(e0 + mm) * KPAD + k4 + kb]);
            v2f a1 = ld_v2f(&Et[(e0 + 16 + mm) * KPAD + k4 + kb]);
            v2f b0 = ld_v2f(&Ft[(i0 + mm) * KPAD + k4 + kb]);
            v2f b1 = ld_v2f(&Ft[(i0 + 16 + mm) * KPAD + k4 + kb]);
            acc[0][0] = wmma_f32(a0, b0, acc[0][0]);
            acc[0][1] = wmma_f32(a0, b1, acc[0][1]);
            acc[1][0] = wmma_f32(a1, b0, acc[1][0]);
            acc[1][1] = wmma_f32(a1, b1, acc[1][1]);
        }
        if (more) {
            float* Ed = smem + (1 - cur) * ABUF;
            stage_chunkA(Ed, Ed + EBUF, tid, ev, fv);
        }
        __syncthreads();
    }

#pragma unroll
    for (int ei = 0; ei < 2; ++ei) {
#pragma unroll
        for (int ii = 0; ii < 2; ++ii) {
            const int ebase = e0 + ei * 16 + half * 8;
            const int icol  = i0 + ii * 16 + mm;
#pragma unroll
            for (int r = 0; r < 8; ++r)
                spec_in[((size_t)g * EE + ebase + r) * II + icol] = acc[ei][ii][r];
        }
    }
}

// ---------------------------------------------------------------------------
// Kernel 2: per-eigen fused MLP (X @ W1 -> relu -> @ W2).
// X, Hb row-major (A-side, K-contiguous already); W1/W2 transposed in LDS.
// Wave w: 2 m-tiles x 4 n-tiles.
// ---------------------------------------------------------------------------
constexpr int PADA = 132;  // row-major A-side: bank = 4m+k, conflict-free
constexpr int PADW = 130;  // transposed B-side: bank = 2n+k, conflict-free

__global__ __launch_bounds__(256) void k_mlp(const float* __restrict__ spec_in,
                                             const float* __restrict__ w1,
                                             const float* __restrict__ b1,
                                             const float* __restrict__ w2,
                                             const float* __restrict__ b2,
                                             float* __restrict__ spec_out) {
    const int e    = blockIdx.x;
    const int tid  = threadIdx.x;
    const int lane = tid & 31;
    const int wave = tid >> 5;

    extern __shared__ float smem[];
    float* X  = smem;                    // [128][PADA]
    float* Wt = X + 128 * PADA;          // [128][PADW]  (n-major, k fast)
    float* Hb = Wt + 128 * PADW;         // [128][PADA]

    // X: gather rows (g, e, :) -> row-major
    for (int idx = tid; idx < 128 * 32; idx += 256) {
        const int row = idx >> 5, c4 = idx & 31;
        float4 v = ((const float4*)(spec_in + ((size_t)row * EE + e) * II))[c4];
        *(float4*)&X[row * PADA + c4 * 4] = v;
    }
    // W1[e]: transpose-scatter -> Wt[n][k]
    {
        const float4* wsrc = (const float4*)(w1 + (size_t)e * II * HH);
        for (int idx = tid; idx < 128 * 32; idx += 256) {
            const int k = idx >> 5, c4 = idx & 31;
            float4 v = wsrc[idx];
            Wt[(c4 * 4 + 0) * PADW + k] = v.x;
            Wt[(c4 * 4 + 1) * PADW + k] = v.y;
            Wt[(c4 * 4 + 2) * PADW + k] = v.z;
            Wt[(c4 * 4 + 3) * PADW + k] = v.w;
        }
    }
    __syncthreads();

    const int mm   = lane & 15;
    const int kb   = (lane >> 4) << 1;
    const int half = lane >> 4;
    const int m0   = (wave & 3) * 32;
    const int n0   = (wave >> 2) * 64;

    v8f acc[2][4] = {};

    // GEMM1: Hid = X @ W1
#pragma unroll 4
    for (int ks = 0; ks < 32; ++ks) {
        const int k4 = ks * 4;
        v2f a0 = ld_v2f(&X[(m0 + mm) * PADA + k4 + kb]);
        v2f a1 = ld_v2f(&X[(m0 + 16 + mm) * PADA + k4 + kb]);
#pragma unroll
        for (int t = 0; t < 4; ++t) {
            v2f b = ld_v2f(&Wt[(n0 + t * 16 + mm) * PADW + k4 + kb]);
            acc[0][t] = wmma_f32(a0, b, acc[0][t]);
            acc[1][t] = wmma_f32(a1, b, acc[1][t]);
        }
    }
    // bias + relu -> Hb (row-major, becomes A of GEMM2)
#pragma unroll
    for (int t = 0; t < 4; ++t) {
        const float bias = b1[e * HH + n0 + t * 16 + mm];
#pragma unroll
        for (int mi = 0; mi < 2; ++mi) {
#pragma unroll
            for (int r = 0; r < 8; ++r) {
                float v = acc[mi][t][r] + bias;
                v = fmaxf(v, 0.f);
                Hb[(m0 + mi * 16 + r + half * 8) * PADA + n0 + t * 16 + mm] = v;
            }
        }
    }
    __syncthreads();   // GEMM1 fully done -> safe to overwrite Wt
    {
        const float4* wsrc = (const float4*)(w2 + (size_t)e * HH * OO);
        for (int idx = tid; idx < 128 * 32; idx += 256) {
            const int k = idx >> 5, c4 = idx & 31;
            float4 v = wsrc[idx];
            Wt[(c4 * 4 + 0) * PADW + k] = v.x;
            Wt[(c4 * 4 + 1) * PADW + k] = v.y;
            Wt[(c4 * 4 + 2) * PADW + k] = v.z;
            Wt[(c4 * 4 + 3) * PADW + k] = v.w;
        }
    }
    __syncthreads();

#pragma unroll
    for (int mi = 0; mi < 2; ++mi)
#pragma unroll
        for (int t = 0; t < 4; ++t) acc[mi][t] = v8f{};

    // GEMM2: out = Hid @ W2
#pragma unroll 4
    for (int ks = 0; ks < 32; ++ks) {
        const int k4 = ks * 4;
        v2f a0 = ld_v2f(&Hb[(m0 + mm) * PADA + k4 + kb]);
        v2f a1 = ld_v2f(&Hb[(m0 + 16 + mm) * PADA + k4 + kb]);
#pragma unroll
        for (int t = 0; t < 4; ++t) {
            v2f b = ld_v2f(&Wt[(n0 + t * 16 + mm) * PADW + k4 + kb]);
            acc[0][t] = wmma_f32(a0, b, acc[0][t]);
            acc[1][t] = wmma_f32(a1, b, acc[1][t]);
        }
    }
#pragma unroll
    for (int t = 0; t < 4; ++t) {
        const float bias = b2[e * OO + n0 + t * 16 + mm];
#pragma unroll
        for (int mi = 0; mi < 2; ++mi) {
#pragma unroll
            for (int r = 0; r < 8; ++r) {
                const int gg = m0 + mi * 16 + r + half * 8;
                spec_out[((size_t)gg * EE + e) * OO + n0 + t * 16 + mm] =
                    acc[mi][t][r] + bias;
            }
        }
    }
}

// ---------------------------------------------------------------------------
// Kernel 3: h = eigen @ spec_out per graph + BN partial stats.
// eigen rows row-major (A-side); spec_out transposed in LDS (B-side).
// Grid: G*8 blocks x 128 n-rows. Wave: 2 m-tiles x 4 n-tiles.
// ---------------------------------------------------------------------------
constexpr int PADE = 68;   // eigen rows [128][68]: bank = 4m+k
constexpr int PADS = 66;   // Bt [128][66]: bank = 2o+k

__global__ __launch_bounds__(256) void k_invproj(const float* __restrict__ eigen,
                                                 const float* __restrict__ spec_out,
                                                 float* __restrict__ h,
                                                 float* __restrict__ gstats) {
    const int g    = blockIdx.x >> 3;
    const int nblk = (blockIdx.x & 7) * 128;
    const int tid  = threadIdx.x;
    const int lane = tid & 31;
    const int wave = tid >> 5;

    extern __shared__ float smem[];
    float* Ae = smem;                  // [128][PADE] eigen rows
    float* Bt = Ae + 128 * PADE;       // [128][PADS] spec_out^T (o-major, e fast)
    float* st = Bt + 128 * PADS;       // [256] partial sum / sumsq

    {
        const float4* esrc = (const float4*)(eigen + ((size_t)g * NN + nblk) * EE);
        for (int idx = tid; idx < 128 * 16; idx += 256) {
            const int row = idx >> 4, c4 = idx & 15;
            *(float4*)&Ae[row * PADE + c4 * 4] = esrc[idx];
        }
        const float4* ssrc = (const float4*)(spec_out + (size_t)g * EE * OO);
        for (int idx = tid; idx < 64 * 32; idx += 256) {
            const int k = idx >> 5, c4 = idx & 31;   // k = e row, c4 = o group
            float4 v = ssrc[idx];
            Bt[(c4 * 4 + 0) * PADS + k] = v.x;
            Bt[(c4 * 4 + 1) * PADS + k] = v.y;
            Bt[(c4 * 4 + 2) * PADS + k] = v.z;
            Bt[(c4 * 4 + 3) * PADS + k] = v.w;
        }
        st[tid] = 0.f;
    }
    __syncthreads();

    const int mm   = lane & 15;
    const int kb   = (lane >> 4) << 1;
    const int half = lane >> 4;
    const int m0   = (wave & 3) * 32;
    const int o0   = (wave >> 2) * 64;

    v8f acc[2][4] = {};

#pragma unroll 4
    for (int ks = 0; ks < 16; ++ks) {   // K = EE = 64
        const int k4 = ks * 4;
        v2f a0 = ld_v2f(&Ae[(m0 + mm) * PADE + k4 + kb]);
        v2f a1 = ld_v2f(&Ae[(m0 + 16 + mm) * PADE + k4 + kb]);
#pragma unroll
        for (int t = 0; t < 4; ++t) {
            v2f b = ld_v2f(&Bt[(o0 + t * 16 + mm) * PADS + k4 + kb]);
            acc[0][t] = wmma_f32(a0, b, acc[0][t]);
            acc[1][t] = wmma_f32(a1, b, acc[1][t]);
        }
    }

    float* hbase = h + ((size_t)g * NN + nblk) * OO;
#pragma unroll
    for (int t = 0; t < 4; ++t) {
        const int oc = o0 + t * 16 + mm;
        float s = 0.f, s2 = 0.f;
#pragma unroll
        for (int mi = 0; mi < 2; ++mi) {
#pragma unroll
            for (int r = 0; r < 8; ++r) {
                const float v = acc[mi][t][r];
                hbase[(size_t)(m0 + mi * 16 + r + half * 8) * OO + oc] = v;
                s  += v;
                s2 += v * v;
            }
        }
        atomicAdd(&st[oc], s);          // ds_add_f32
        atomicAdd(&st[128 + oc], s2);
    }
    __syncthreads();
    atomicAdd(&gstats[tid], st[tid]);   // global_atomic_add_f32
}

// ---------------------------------------------------------------------------
// Small kernels: stats init, BN fold, epilogue.
// ---------------------------------------------------------------------------
__global__ void k_zero(float* __restrict__ p, int n) {
    const int i = blockIdx.x * blockDim.x + threadIdx.x;
    if (i < n) p[i] = 0.f;
}

__global__ void k_bnfinal(const float* __restrict__ gstats,
                          const float* __restrict__ gamma,
                          const float* __restrict__ beta,
                          float* __restrict__ params) {
    const int o = threadIdx.x;
    if (o < OO) {
        const float invM = 1.f / (float)(GG * NN);
        const float mean = gstats[o] * invM;
        const float var  = gstats[128 + o] * invM - mean * mean;
        const float rstd = rsqrtf(var + 1e-5f);
        const float sc   = gamma[o] * rstd;
        params[o]        = sc;
        params[128 + o]  = beta[o] - mean * sc;
    }
}

__global__ __launch_bounds__(256) void k_epilogue(const float* __restrict__ feature,
                                                  const float* __restrict__ h,
                                                  const float* __restrict__ params,
                                                  float* __restrict__ out) {
    __shared__ float sc[OO], sh[OO];
    if (threadIdx.x < OO) {
        sc[threadIdx.x] = params[threadIdx.x];
        sh[threadIdx.x] = params[128 + threadIdx.x];
    }
    __syncthreads();
    const size_t total4 = (size_t)GG * NN * OO / 4;
    for (size_t i = (size_t)blockIdx.x * blockDim.x + threadIdx.x; i < total4;
         i += (size_t)gridDim.x * blockDim.x) {
        const int c = (int)((i * 4) & (OO - 1));
        const float4 f  = ((const float4*)feature)[i];
        const float4 hv = ((const float4*)h)[i];
        float4 o4;
        o4.x = f.x + fmaxf(0.f, hv.x * sc[c + 0] + sh[c + 0]);
        o4.y = f.y + fmaxf(0.f, hv.y * sc[c + 1] + sh[c + 1]);
        o4.z = f.z + fmaxf(0.f, hv.z * sc[c + 2] + sh[c + 2]);
        o4.w = f.w + fmaxf(0.f, hv.w * sc[c + 3] + sh[c + 3]);
        ((float4*)out)[i] = o4;
    }
}

// ---------------------------------------------------------------------------
extern "C" void kernel_launch(void* const* d_in, const int* in_sizes, int n_in,
                              void* d_out, int out_size, void* d_ws, size_t ws_size,
                              hipStream_t stream) {
    (void)in_sizes; (void)n_in; (void)out_size; (void)ws_size;
    const float* feature = (const float*)d_in[0];
    const float* eigen   = (const float*)d_in[1];
    const float* w1      = (const float*)d_in[2];
    const float* b1      = (const float*)d_in[3];
    const float* w2      = (const float*)d_in[4];
    const float* b2      = (const float*)d_in[5];
    const float* gamma   = (const float*)d_in[6];
    const float* beta    = (const float*)d_in[7];
    float* out = (float*)d_out;

    // Workspace layout (floats): spec_in | spec_out | h | gstats(256) | params(256)
    float* spec_in  = (float*)d_ws;
    float* spec_out = spec_in  + (size_t)GG * EE * II;
    float* hbuf     = spec_out + (size_t)GG * EE * OO;
    float* gstats   = hbuf     + (size_t)GG * NN * OO;
    float* params   = gstats + 256;

    k_zero<<<1, 256, 0, stream>>>(gstats, 256);

    const size_t smemA = (size_t)2 * ABUF * sizeof(float);                   // ~51 KB
    k_spec_in<<<GG, 256, smemA, stream>>>(feature, eigen, spec_in);

    const size_t smemB = (size_t)(128 * PADA * 2 + 128 * PADW) * sizeof(float); // ~197 KB
    k_mlp<<<EE, 256, smemB, stream>>>(spec_in, w1, b1, w2, b2, spec_out);

    const size_t smemC = (size_t)(128 * PADE + 128 * PADS + 256) * sizeof(float);
    k_invproj<<<GG * 8, 256, smemC, stream>>>(eigen, spec_out, hbuf, gstats);

    k_bnfinal<<<1, 128, 0, stream>>>(gstats, gamma, beta, params);
    k_epilogue<<<2048, 256, 0, stream>>>(feature, hbuf, params, out);
}